// PerceiverAttentionLayer_54571854463453
// MI455X (gfx1250) — compile-verified
//
#include <hip/hip_runtime.h>
#include <math.h>

typedef __attribute__((ext_vector_type(2))) float v2f;
typedef __attribute__((ext_vector_type(8))) float v8f;

typedef __attribute__((ext_vector_type(4))) unsigned int u32x4_t;
typedef __attribute__((ext_vector_type(4))) int i32x4_t;
typedef __attribute__((ext_vector_type(8))) int i32x8_t;

#define DIMX 1024
#define HEADS 16
#define DIM_HEAD 64
#define NB 16
#define NF 4096
#define NQ 64
#define NKV 4160          // NF + NQ
#define SCALE_Q 0.125f    // 64^-0.5
#define EPS_LN 1e-5f
#define NEG_INF (-__builtin_inff())

#if defined(__has_builtin)
#if __has_builtin(__builtin_amdgcn_tensor_load_to_lds) && \
    __has_builtin(__builtin_amdgcn_s_wait_tensorcnt)
#define HAVE_TDM 1
#endif
#endif

#ifdef HAVE_TDM
// LDS byte offset of a __shared__ object: low 32 bits of the flat address
// (LDS aperture lives entirely in the high dword).
__device__ __forceinline__ unsigned int lds_off(const void* p)
{
    return (unsigned int)(uintptr_t)p;
}

// Issue a TDM 2D tile load: tileH rows x tileW f32 elements from a row-major
// tensor with rowStrideElems f32 per row, into LDS at ldsOff. Optional LDS
// padding (pad_amount code 0 == 1 dword) every 2^(padIntervalCode+1) dwords.
// D# layout per CDNA5 ISA ch.8 (group0: count/lds/global/type, group1: dims).
__device__ __forceinline__ void tdm_load_tile_2d(
    const float* gsrc, unsigned int ldsOff,
    unsigned int tileW, unsigned int tileH,
    unsigned int rowStrideElems, unsigned int tensorRows,
    unsigned int padEn, unsigned int padIntervalCode)
{
    unsigned long long ga = (unsigned long long)(uintptr_t)gsrc;
    u32x4_t g0;
    g0[0] = 1u;                                            // count=1 (valid user D#)
    g0[1] = ldsOff;                                        // lds_addr
    g0[2] = (unsigned int)ga;                              // global_addr[31:0]
    g0[3] = (unsigned int)((ga >> 32) & 0x01FFFFFFull)     // global_addr[56:32]
          | (2u << 30);                                    // type = 2 ("image")
    i32x8_t g1;
    g1[0] = (int)((2u << 16)                               // data_size = 4 bytes
          | (padEn << 20)                                  // pad_enable
          | ((padIntervalCode & 7u) << 22));               // pad_interval; pad_amount=0 (1 dword)
    g1[1] = (int)((rowStrideElems & 0xFFFFu) << 16);       // tensor_dim0[15:0]
    g1[2] = (int)(((rowStrideElems >> 16) & 0xFFFFu)       // tensor_dim0[31:16]
          | ((tensorRows & 0xFFFFu) << 16));               // tensor_dim1[15:0]
    g1[3] = (int)(((tensorRows >> 16) & 0xFFFFu)           // tensor_dim1[31:16]
          | ((tileW & 0xFFFFu) << 16));                    // tile_dim0
    g1[4] = (int)(tileH & 0xFFFFu);                        // tile_dim1 (tile_dim2=0)
    g1[5] = (int)rowStrideElems;                           // tensor_dim0_stride[31:0]
    g1[6] = 0;                                             // stride[47:32], dim1_stride[15:0]
    g1[7] = 0;
    i32x4_t gz = {0, 0, 0, 0};                             // groups 2/3 unused (2D)
#if __clang_major__ >= 23
    i32x8_t gz8 = {0, 0, 0, 0, 0, 0, 0, 0};
    __builtin_amdgcn_tensor_load_to_lds(g0, g1, gz, gz, gz8, 0);
#else
    __builtin_amdgcn_tensor_load_to_lds(g0, g1, gz, gz, 0);
#endif
}
#endif // HAVE_TDM

// ---------------------------------------------------------------------------
// Per-row LayerNorm statistics (mu, 1/sqrt(var+eps)) for features and latents.
// ---------------------------------------------------------------------------
__global__ __launch_bounds__(256)
void stats_kernel(const float* __restrict__ feat, const float* __restrict__ lat,
                  float* __restrict__ mu, float* __restrict__ rinv)
{
    int row = blockIdx.x;  // 0..66559 (65536 feature rows, then 1024 latent rows)
    const float* src = (row < NB * NF) ? (feat + (size_t)row * DIMX)
                                       : (lat + (size_t)(row - NB * NF) * DIMX);
    int tid = threadIdx.x;
    float s = 0.f, sq = 0.f;
    for (int i = tid; i < DIMX; i += 256) {
        float v = src[i];
        s += v; sq += v * v;
    }
    for (int o = 16; o > 0; o >>= 1) {
        s  += __shfl_down(s, o, 32);
        sq += __shfl_down(sq, o, 32);
    }
    __shared__ float red[16];
    int w = tid >> 5, l = tid & 31;
    if (l == 0) { red[w * 2] = s; red[w * 2 + 1] = sq; }
    __syncthreads();
    if (tid == 0) {
        float S = 0.f, SQ = 0.f;
        for (int i = 0; i < 8; ++i) { S += red[i * 2]; SQ += red[i * 2 + 1]; }
        float m = S * (1.0f / DIMX);
        float var = SQ * (1.0f / DIMX) - m * m;
        mu[row] = m;
        rinv[row] = rsqrtf(var + EPS_LN);
    }
}

// ---------------------------------------------------------------------------
// Fold gamma into weights: Wf[v] = g_v[k] * W_v[k][n]
// variants: 0:Wk(gm) 1:Wk(gl) 2:Wv(gm) 3:Wv(gl) 4:Wq(gl)
// ---------------------------------------------------------------------------
__global__ __launch_bounds__(256)
void fold_kernel(const float* __restrict__ Wq, const float* __restrict__ Wk,
                 const float* __restrict__ Wv, const float* __restrict__ gm,
                 const float* __restrict__ gl, float* __restrict__ Wf)
{
    int v = blockIdx.y;
    int idx = blockIdx.x * 256 + threadIdx.x;   // 0..1M-1
    int k = idx >> 10;
    const float* W = (v < 2) ? Wk : (v < 4) ? Wv : Wq;
    const float* g = (v == 0 || v == 2) ? gm : gl;
    Wf[(size_t)v * DIMX * DIMX + idx] = g[k] * W[idx];
}

// c1[v][n] = sum_k g_v[k] W_v[k][n];  c2[v][n] = sum_k b_v[k] W_v[k][n]
__global__ __launch_bounds__(256)
void colsum_kernel(const float* __restrict__ Wq, const float* __restrict__ Wk,
                   const float* __restrict__ Wv, const float* __restrict__ gm,
                   const float* __restrict__ gl, const float* __restrict__ bm,
                   const float* __restrict__ bl, float* __restrict__ c1,
                   float* __restrict__ c2)
{
    int v = blockIdx.y;
    int n = blockIdx.x * 256 + threadIdx.x;
    const float* W = (v < 2) ? Wk : (v < 4) ? Wv : Wq;
    const float* g = (v == 0 || v == 2) ? gm : gl;
    const float* b = (v == 0 || v == 2) ? bm : bl;
    float a1 = 0.f, a2 = 0.f;
    for (int k = 0; k < DIMX; ++k) {
        float w = W[(size_t)k * DIMX + n];
        a1 += g[k] * w;
        a2 += b[k] * w;
    }
    c1[v * DIMX + n] = a1;
    c2[v * DIMX + n] = a2;
}

// ---------------------------------------------------------------------------
// Tiled GEMM with LN-fold epilogue:  out = alpha*( rinv*(A@Wf) - rinv*mu*c1 + c2 )
// 128x128 tile per 256-thread block (8 waves, each 32x64), f32 WMMA 16x16x4.
// TDM double-buffered LDS staging when available.
// mode: output address mapping (0 direct, 1 feat->K/V head-major,
//       2 lat->K/V head-major, 3 lat->Q head-major)
// ---------------------------------------------------------------------------
__global__ __launch_bounds__(256)
void gemm_ln_kernel(const float* __restrict__ A, const float* __restrict__ W,
                    const float* __restrict__ c1, const float* __restrict__ c2,
                    const float* __restrict__ mu, const float* __restrict__ rinv,
                    float* __restrict__ out, float alpha, int mode, int lnOn)
{
#ifdef HAVE_TDM
    __shared__ __align__(16) float As[2][128][17];  // 17 -> conflict-free column reads
    __shared__ __align__(16) float Bs[2][16][128];
#else
    __shared__ __align__(16) float As[1][128][17];
    __shared__ __align__(16) float Bs[1][16][128];
#endif

    int tid  = threadIdx.x;
    int lane = tid & 31, w = tid >> 5;
    int half = lane >> 4, l = lane & 15;
    int mBase = (w >> 1) * 32;   // 4 M-groups of 32 rows
    int nBase = (w & 1) * 64;    // 2 N-groups of 64 cols
    size_t rowBase = (size_t)blockIdx.x * 128;
    int colBase = blockIdx.y * 128;

    v8f c[2][4];
    for (int mi = 0; mi < 2; ++mi)
        for (int ni = 0; ni < 4; ++ni)
            for (int e = 0; e < 8; ++e) c[mi][ni][e] = 0.f;

    const float* Ap = A + rowBase * DIMX;

#ifdef HAVE_TDM
    // prologue: DMA tile 0 into buffer 0 (wave 0 drives the TDM)
    if (w == 0) {
        tdm_load_tile_2d(Ap,           lds_off(&As[0][0][0]), 16, 128, DIMX, 128, 1, 3);
        tdm_load_tile_2d(W + colBase,  lds_off(&Bs[0][0][0]), 128, 16, DIMX, DIMX, 0, 0);
    }
#endif

    for (int k0 = 0; k0 < DIMX; k0 += 16) {
#ifdef HAVE_TDM
        int cur = (k0 >> 4) & 1;
        if (w == 0) {
            if (k0 + 16 < DIMX) {
                int nxt = cur ^ 1;
                tdm_load_tile_2d(Ap + (k0 + 16),
                                 lds_off(&As[nxt][0][0]), 16, 128, DIMX, 128, 1, 3);
                tdm_load_tile_2d(W + (size_t)(k0 + 16) * DIMX + colBase,
                                 lds_off(&Bs[nxt][0][0]), 128, 16, DIMX,
                                 (unsigned)(DIMX - (k0 + 16)), 0, 0);
                // 2 new loads in flight; wait until the 2 older (current-buffer)
                // transfers are complete (TDM is in-order per wave).
                __builtin_amdgcn_s_wait_tensorcnt(2);
            } else {
                __builtin_amdgcn_s_wait_tensorcnt(0);
            }
        }
        __syncthreads();
#else
        const int cur = 0;
        // stage A tile 128x16
        for (int i = 0; i < 2; ++i) {
            int id = tid + i * 256;
            int r = id >> 2, c4 = (id & 3) * 4;
            const float4 va = *(const float4*)(Ap + (size_t)r * DIMX + k0 + c4);
            As[0][r][c4 + 0] = va.x; As[0][r][c4 + 1] = va.y;
            As[0][r][c4 + 2] = va.z; As[0][r][c4 + 3] = va.w;
        }
        // stage B tile 16x128
        for (int i = 0; i < 2; ++i) {
            int id = tid + i * 256;
            int r = id >> 5, c4 = (id & 31) * 4;
            *(float4*)&Bs[0][r][c4] =
                *(const float4*)(W + (size_t)(k0 + r) * DIMX + colBase + c4);
        }
        if (k0 + 16 < DIMX) {   // prefetch next k-tile -> global_prefetch_b8
            __builtin_prefetch(Ap + (size_t)(tid >> 1) * DIMX + k0 + 16, 0, 1);
            __builtin_prefetch(W + (size_t)(k0 + 16 + (tid >> 5)) * DIMX + colBase + (tid & 31) * 4, 0, 1);
        }
        __syncthreads();
#endif

        for (int kk = 0; kk < 16; kk += 4) {
            v2f a[2], b[4];
            for (int mi = 0; mi < 2; ++mi) {
                int r = mBase + mi * 16 + l;
                a[mi].x = As[cur][r][kk + 2 * half];
                a[mi].y = As[cur][r][kk + 2 * half + 1];
            }
            for (int ni = 0; ni < 4; ++ni) {
                int cc = nBase + ni * 16 + l;
                b[ni].x = Bs[cur][kk + 2 * half][cc];
                b[ni].y = Bs[cur][kk + 2 * half + 1][cc];
            }
            for (int mi = 0; mi < 2; ++mi)
                for (int ni = 0; ni < 4; ++ni)
                    c[mi][ni] = __builtin_amdgcn_wmma_f32_16x16x4_f32(
                        false, a[mi], false, b[ni], (short)0, c[mi][ni], false, false);
        }
        __syncthreads();
    }

    // epilogue: LN fold + scatter per mode
    for (int mi = 0; mi < 2; ++mi) {
        float rv[8], mm[8];
        if (lnOn) {
            for (int j = 0; j < 8; ++j) {
                size_t ar = rowBase + mBase + mi * 16 + j + 8 * half;
                rv[j] = rinv[ar];
                mm[j] = mu[ar];
            }
        }
        for (int ni = 0; ni < 4; ++ni) {
            int n = colBase + nBase + ni * 16 + l;
            float c1n = lnOn ? c1[n] : 0.f;
            float c2n = lnOn ? c2[n] : 0.f;
            for (int j = 0; j < 8; ++j) {
                size_t ar = rowBase + mBase + mi * 16 + j + 8 * half;
                float v = c[mi][ni][j];
                if (lnOn) v = rv[j] * v - rv[j] * mm[j] * c1n + c2n;
                v *= alpha;
                size_t addr;
                if (mode == 0) {                       // direct (final @Wo)
                    addr = ar * DIMX + n;
                } else if (mode == 1) {                // features -> K/V [b][h][t][d]
                    size_t bb = ar >> 12, t = ar & 4095;
                    size_t h = (size_t)(n >> 6), d = (size_t)(n & 63);
                    addr = ((bb * HEADS + h) * NKV + t) * DIM_HEAD + d;
                } else if (mode == 2) {                // latents -> K/V tail rows
                    size_t bb = ar >> 6, q = ar & 63;
                    size_t h = (size_t)(n >> 6), d = (size_t)(n & 63);
                    addr = ((bb * HEADS + h) * NKV + NF + q) * DIM_HEAD + d;
                } else {                               // latents -> Q [b][h][q][d]
                    size_t bb = ar >> 6, q = ar & 63;
                    size_t h = (size_t)(n >> 6), d = (size_t)(n & 63);
                    addr = ((bb * HEADS + h) * NQ + q) * DIM_HEAD + d;
                }
                out[addr] = v;
            }
        }
    }
}

// ---------------------------------------------------------------------------
// Attention: one block per (b,h). Two-pass masked softmax, WMMA for QK^T and PV.
// K/V/Q tiles DMA'd into padded LDS rows by the TDM when available.
// ---------------------------------------------------------------------------
__global__ __launch_bounds__(256)
void attn_kernel(const float* __restrict__ Q, const float* __restrict__ K,
                 const float* __restrict__ V, const unsigned char* __restrict__ masks,
                 float* __restrict__ attn)
{
    __shared__ __align__(16) float qs[64][65];
    __shared__ __align__(16) float ks[64][65];
    __shared__ __align__(16) float vs[64][65];
    __shared__ __align__(16) float ss[64][65];
    __shared__ float ms[64];
    __shared__ float red[4][64];
    __shared__ float rmax[64];
    __shared__ float rsum[64];

    int bh = blockIdx.x;           // b*HEADS + h
    int b = bh >> 4;
    int tid = threadIdx.x;
    int lane = tid & 31, w = tid >> 5;
    int half = lane >> 4, l = lane & 15;
    int mBase = (w >> 1) * 16;     // 4 q-row groups of 16
    int nBase = (w & 1) * 32;      // 2 col groups of 32

    const float* Qb = Q + (size_t)bh * NQ * DIM_HEAD;
    const float* Kb = K + (size_t)bh * NKV * DIM_HEAD;
    const float* Vb = V + (size_t)bh * NKV * DIM_HEAD;

#ifdef HAVE_TDM
    if (w == 0) {
        tdm_load_tile_2d(Qb, lds_off(&qs[0][0]), 64, 64, 64, 64, 1, 5); // pad to 65
        __builtin_amdgcn_s_wait_tensorcnt(0);
    }
#else
    for (int i = 0; i < 16; ++i) {
        int idx = tid + i * 256;
        qs[idx >> 6][idx & 63] = Qb[idx];
    }
#endif
    __syncthreads();

    int srow = tid & 63;           // row this thread scans in softmax passes
    int sgrp = tid >> 6;           // col group 0..3 (16 cols each)

    // ---------------- pass 1: masked row max ----------------
    float pmax = NEG_INF;
    for (int ch = 0; ch < 65; ++ch) {
        int t0 = ch * 64;
#ifdef HAVE_TDM
        if (w == 0) {
            tdm_load_tile_2d(Kb + (size_t)t0 * DIM_HEAD, lds_off(&ks[0][0]),
                             64, 64, 64, 64, 1, 5);
            __builtin_amdgcn_s_wait_tensorcnt(0);
        }
#else
        for (int i = 0; i < 16; ++i) {
            int idx = tid + i * 256;
            ks[idx >> 6][idx & 63] = Kb[(size_t)t0 * DIM_HEAD + idx];
        }
#endif
        if (tid < 64) {
            int t = t0 + tid;
            ms[tid] = (t < NF) ? (masks[(size_t)b * NF + t] ? 1.f : 0.f) : 1.f;
        }
        __syncthreads();

        v8f sc[2];
        for (int ni = 0; ni < 2; ++ni)
            for (int e = 0; e < 8; ++e) sc[ni][e] = 0.f;
        for (int kk = 0; kk < 64; kk += 4) {
            v2f a, bb[2];
            a.x = qs[mBase + l][kk + 2 * half];
            a.y = qs[mBase + l][kk + 2 * half + 1];
            for (int ni = 0; ni < 2; ++ni) {
                int j = nBase + ni * 16 + l;
                bb[ni].x = ks[j][kk + 2 * half];
                bb[ni].y = ks[j][kk + 2 * half + 1];
            }
            for (int ni = 0; ni < 2; ++ni)
                sc[ni] = __builtin_amdgcn_wmma_f32_16x16x4_f32(
                    false, a, false, bb[ni], (short)0, sc[ni], false, false);
        }
        for (int ni = 0; ni < 2; ++ni)
            for (int j = 0; j < 8; ++j)
                ss[mBase + j + 8 * half][nBase + ni * 16 + l] = sc[ni][j];
        __syncthreads();
        for (int cc = 0; cc < 16; ++cc) {
            int col = sgrp * 16 + cc;
            float v = ms[col] > 0.f ? ss[srow][col] : NEG_INF;
            pmax = fmaxf(pmax, v);
        }
        __syncthreads();
    }
    red[sgrp][srow] = pmax;
    __syncthreads();
    if (tid < 64)
        rmax[tid] = fmaxf(fmaxf(red[0][tid], red[1][tid]),
                          fmaxf(red[2][tid], red[3][tid]));
    __syncthreads();

    // ---------------- pass 2: exp + accumulate ----------------
    v8f oc[2];
    for (int ni = 0; ni < 2; ++ni)
        for (int e = 0; e < 8; ++e) oc[ni][e] = 0.f;
    float psum = 0.f;

    for (int ch = 0; ch < 65; ++ch) {
        int t0 = ch * 64;
#ifdef HAVE_TDM
        if (w == 0) {
            tdm_load_tile_2d(Kb + (size_t)t0 * DIM_HEAD, lds_off(&ks[0][0]),
                             64, 64, 64, 64, 1, 5);
            tdm_load_tile_2d(Vb + (size_t)t0 * DIM_HEAD, lds_off(&vs[0][0]),
                             64, 64, 64, 64, 1, 5);
            __builtin_amdgcn_s_wait_tensorcnt(0);
        }
#else
        for (int i = 0; i < 16; ++i) {
            int idx = tid + i * 256;
            ks[idx >> 6][idx & 63] = Kb[(size_t)t0 * DIM_HEAD + idx];
            vs[idx >> 6][idx & 63] = Vb[(size_t)t0 * DIM_HEAD + idx];
        }
#endif
        if (tid < 64) {
            int t = t0 + tid;
            ms[tid] = (t < NF) ? (masks[(size_t)b * NF + t] ? 1.f : 0.f) : 1.f;
        }
        __syncthreads();

        v8f sc[2];
        for (int ni = 0; ni < 2; ++ni)
            for (int e = 0; e < 8; ++e) sc[ni][e] = 0.f;
        for (int kk = 0; kk < 64; kk += 4) {
            v2f a, bb[2];
            a.x = qs[mBase + l][kk + 2 * half];
            a.y = qs[mBase + l][kk + 2 * half + 1];
            for (int ni = 0; ni < 2; ++ni) {
                int j = nBase + ni * 16 + l;
                bb[ni].x = ks[j][kk + 2 * half];
                bb[ni].y = ks[j][kk + 2 * half + 1];
            }
            for (int ni = 0; ni < 2; ++ni)
                sc[ni] = __builtin_amdgcn_wmma_f32_16x16x4_f32(
                    false, a, false, bb[ni], (short)0, sc[ni], false, false);
        }
        for (int ni = 0; ni < 2; ++ni)
            for (int j = 0; j < 8; ++j)
                ss[mBase + j + 8 * half][nBase + ni * 16 + l] = sc[ni][j];
        __syncthreads();

        float m = rmax[srow];
        for (int cc = 0; cc < 16; ++cc) {
            int col = sgrp * 16 + cc;
            float p = ms[col] > 0.f ? __expf(ss[srow][col] - m) : 0.f;
            ss[srow][col] = p;
            psum += p;
        }
        __syncthreads();

        for (int kk = 0; kk < 64; kk += 4) {
            v2f a, bb[2];
            a.x = ss[mBase + l][kk + 2 * half];
            a.y = ss[mBase + l][kk + 2 * half + 1];
            for (int ni = 0; ni < 2; ++ni) {
                int d = nBase + ni * 16 + l;
                bb[ni].x = vs[kk + 2 * half][d];
                bb[ni].y = vs[kk + 2 * half + 1][d];
            }
            for (int ni = 0; ni < 2; ++ni)
                oc[ni] = __builtin_amdgcn_wmma_f32_16x16x4_f32(
                    false, a, false, bb[ni], (short)0, oc[ni], false, false);
        }
        __syncthreads();
    }
    red[sgrp][srow] = psum;
    __syncthreads();
    if (tid < 64)
        rsum[tid] = red[0][tid] + red[1][tid] + red[2][tid] + red[3][tid];
    __syncthreads();

    int h = bh & 15;
    for (int ni = 0; ni < 2; ++ni) {
        for (int j = 0; j < 8; ++j) {
            int q = mBase + j + 8 * half;
            int d = nBase + ni * 16 + l;
            attn[((size_t)b * NQ + q) * DIMX + h * DIM_HEAD + d] = oc[ni][j] / rsum[q];
        }
    }
}

// ---------------------------------------------------------------------------
extern "C" void kernel_launch(void* const* d_in, const int* in_sizes, int n_in,
                              void* d_out, int out_size, void* d_ws, size_t ws_size,
                              hipStream_t stream)
{
    const float* features = (const float*)d_in[0];
    const unsigned char* masks = (const unsigned char*)d_in[1];
    const float* latents = (const float*)d_in[2];
    const float* gm = (const float*)d_in[3];
    const float* bm = (const float*)d_in[4];
    const float* gl = (const float*)d_in[5];
    const float* bl = (const float*)d_in[6];
    const float* Wq = (const float*)d_in[7];
    const float* Wk = (const float*)d_in[8];
    const float* Wv = (const float*)d_in[9];
    const float* Wo = (const float*)d_in[10];
    float* out = (float*)d_out;

    float* ws = (float*)d_ws;
    size_t off = 0;
    float* Kbuf = ws + off; off += (size_t)NB * HEADS * NKV * DIM_HEAD;  // 272.6 MB
    float* Vbuf = ws + off; off += (size_t)NB * HEADS * NKV * DIM_HEAD;  // 272.6 MB
    float* Qbuf = ws + off; off += (size_t)NB * HEADS * NQ * DIM_HEAD;   // 4 MB
    float* attn = ws + off; off += (size_t)NB * NQ * DIMX;               // 4 MB
    float* Wf   = ws + off; off += (size_t)5 * DIMX * DIMX;              // 20 MB
    float* c1   = ws + off; off += (size_t)5 * DIMX;
    float* c2   = ws + off; off += (size_t)5 * DIMX;
    float* muA  = ws + off; off += (size_t)(NB * NF + NB * NQ);
    float* rvA  = ws + off; off += (size_t)(NB * NF + NB * NQ);

    stats_kernel<<<NB * NF + NB * NQ, 256, 0, stream>>>(features, latents, muA, rvA);
    fold_kernel<<<dim3(DIMX * DIMX / 256, 5), 256, 0, stream>>>(Wq, Wk, Wv, gm, gl, Wf);
    colsum_kernel<<<dim3(DIMX / 256, 5), 256, 0, stream>>>(Wq, Wk, Wv, gm, gl, bm, bl, c1, c2);

    // K = LN(kv_input) @ Wk   (feature rows + latent rows)
    gemm_ln_kernel<<<dim3(512, 8), 256, 0, stream>>>(
        features, Wf + 0 * DIMX * DIMX, c1 + 0 * DIMX, c2 + 0 * DIMX,
        muA, rvA, Kbuf, 1.f, 1, 1);
    gemm_ln_kernel<<<dim3(8, 8), 256, 0, stream>>>(
        latents, Wf + 1 * (size_t)DIMX * DIMX, c1 + 1 * DIMX, c2 + 1 * DIMX,
        muA + NB * NF, rvA + NB * NF, Kbuf, 1.f, 2, 1);
    // V
    gemm_ln_kernel<<<dim3(512, 8), 256, 0, stream>>>(
        features, Wf + 2 * (size_t)DIMX * DIMX, c1 + 2 * DIMX, c2 + 2 * DIMX,
        muA, rvA, Vbuf, 1.f, 1, 1);
    gemm_ln_kernel<<<dim3(8, 8), 256, 0, stream>>>(
        latents, Wf + 3 * (size_t)DIMX * DIMX, c1 + 3 * DIMX, c2 + 3 * DIMX,
        muA + NB * NF, rvA + NB * NF, Vbuf, 1.f, 2, 1);
    // Q (scaled)
    gemm_ln_kernel<<<dim3(8, 8), 256, 0, stream>>>(
        latents, Wf + 4 * (size_t)DIMX * DIMX, c1 + 4 * DIMX, c2 + 4 * DIMX,
        muA + NB * NF, rvA + NB * NF, Qbuf, SCALE_Q, 3, 1);

    attn_kernel<<<NB * HEADS, 256, 0, stream>>>(Qbuf, Kbuf, Vbuf, masks, attn);

    // final projection: attn @ Wo -> out
    gemm_ln_kernel<<<dim3(8, 8), 256, 0, stream>>>(
        attn, Wo, nullptr, nullptr, nullptr, nullptr, out, 1.f, 0, 0);
}